// SEAS_40956808135232
// MI455X (gfx1250) — compile-verified
//
#include <hip/hip_runtime.h>
#include <hip/hip_bf16.h>
#include <stdint.h>

// ---- static config mirroring the reference ----
#define BIMG      8
#define NPROP     2048
#define NTHREADS  1024
#define DETS      100
#define IMG_W     1344.0f
#define IMG_H     800.0f
#define SCORE_T   0.5f
#define NMS_T     0.4f
#define BBOX_CLIPV 4.135166556742356f   // log(1000/16)
#define MIN_SZ    0.01f
#define NEGV     -1e30f

__device__ __forceinline__ unsigned f2ord(float f) {
    unsigned u = __float_as_uint(f);
    return (u & 0x80000000u) ? ~u : (u | 0x80000000u);
}
__device__ __forceinline__ float ord2f(unsigned ou) {
    unsigned u = (ou & 0x80000000u) ? (ou ^ 0x80000000u) : ~ou;
    return __uint_as_float(u);
}
__device__ __forceinline__ unsigned ballot32(bool p) {
#if __has_builtin(__builtin_amdgcn_ballot_w32)
    return __builtin_amdgcn_ballot_w32(p);
#else
    return (unsigned)__ballot(p);
#endif
}

__global__ __launch_bounds__(NTHREADS)
void SEAS_postprocess_kernel(const float* __restrict__ cls,    // [B,N,2]
                             const float* __restrict__ regs,   // [B,N,4]
                             const float* __restrict__ qlt,    // [B,N,1]
                             const float* __restrict__ props,  // [B,N,4]
                             float* __restrict__ outBoxes,     // [B,100,4]
                             float* __restrict__ outScores,    // [B,100]
                             int*   __restrict__ outClasses,   // [B,100]
                             unsigned char* __restrict__ outValid) // [B,100]
{
    // LDS layout (~48.3 KB): proposals staged by async DMA, then decoded in place
    __shared__ float               sBox[NPROP * 4];   // 32 KB : props -> decoded boxes
    __shared__ unsigned long long  sKey[NPROP];       // 16 KB : (ordered score << 32)|(2047-i)
    __shared__ unsigned            sSup[NPROP / 32];  // 256 B : suppression bitmask
    __shared__ int sPick, sCur, sCount;

    const int b = blockIdx.x;
    const int t = threadIdx.x;

    // ---- 1) Async DMA: stage this image's proposal boxes (32 KB) into LDS ----
    // gfx1250 path: global_load_async_to_lds_b128 per lane, tracked by ASYNCcnt.
    for (int e = 0; e < 2; ++e) {
        const int i = t + e * NTHREADS;
        unsigned long long ga = (unsigned long long)(props + ((size_t)b * NPROP + i) * 4);
        unsigned la = (unsigned)(size_t)&sBox[i * 4];   // low 32 bits of flat ptr = LDS offset
        asm volatile("global_load_async_to_lds_b128 %0, %1, off"
                     :: "v"(la), "v"(ga) : "memory");
    }

    // Warm L2 for the register-path loads of the far half (overlaps with DMA)
    __builtin_prefetch(cls  + ((size_t)b * NPROP + t + NTHREADS) * 2, 0, 0);
    __builtin_prefetch(regs + ((size_t)b * NPROP + t + NTHREADS) * 4, 0, 0);

    // ---- 2) Register-path loads of cls/regs/qlt (each used exactly once) ----
    float c0[2], c1[2], r0[2], r1[2], r2[2], r3[2], q[2];
    for (int e = 0; e < 2; ++e) {
        const size_t i = (size_t)b * NPROP + (t + e * NTHREADS);
        c0[e] = cls[i * 2 + 0];
        c1[e] = cls[i * 2 + 1];
        r0[e] = regs[i * 4 + 0];
        r1[e] = regs[i * 4 + 1];
        r2[e] = regs[i * 4 + 2];
        r3[e] = regs[i * 4 + 3];
        q[e]  = qlt[i];
    }

    if (t < NPROP / 32) sSup[t] = 0u;
    if (t == 0) { sCur = 0; sCount = 0; }

    asm volatile("s_wait_asynccnt 0x0" ::: "memory");   // my waves' DMA done
    __syncthreads();                                    // everyone's DMA visible

    // ---- 3) Decode + clip (in place over staged proposals) + score/key ----
    for (int e = 0; e < 2; ++e) {
        const int i = t + e * NTHREADS;
        const float x1 = sBox[i * 4 + 0], y1 = sBox[i * 4 + 1];
        const float x2 = sBox[i * 4 + 2], y2 = sBox[i * 4 + 3];
        const float w  = x2 - x1,   h  = y2 - y1;
        const float cx = x1 + 0.5f * w, cy = y1 + 0.5f * h;
        const float dx = r0[e] / 10.0f;
        const float dy = r1[e] / 10.0f;
        const float dw = fminf(r2[e] / 5.0f, BBOX_CLIPV);
        const float dh = fminf(r3[e] / 5.0f, BBOX_CLIPV);
        const float pcx = dx * w + cx, pcy = dy * h + cy;
        const float pw  = __expf(dw) * w, ph = __expf(dh) * h;
        const float nx1 = fminf(fmaxf(pcx - 0.5f * pw, 0.0f), IMG_W);
        const float ny1 = fminf(fmaxf(pcy - 0.5f * ph, 0.0f), IMG_H);
        const float nx2 = fminf(fmaxf(pcx + 0.5f * pw, 0.0f), IMG_W);
        const float ny2 = fminf(fmaxf(pcy + 0.5f * ph, 0.0f), IMG_H);
        sBox[i * 4 + 0] = nx1; sBox[i * 4 + 1] = ny1;
        sBox[i * 4 + 2] = nx2; sBox[i * 4 + 3] = ny2;

        const float raw = 1.0f / (1.0f + __expf(c0[e] - c1[e]));   // softmax fg (C=2)
        const float sc  = raw * (1.0f / (1.0f + __expf(-q[e])));   // quality-weighted
        const bool valid = ((nx2 - nx1) >= MIN_SZ) && ((ny2 - ny1) >= MIN_SZ) &&
                           (raw > SCORE_T);
        const float s = valid ? sc : NEGV;
        // descending sort key; secondary key makes smaller original index win ties
        sKey[i] = ((unsigned long long)f2ord(s) << 32) |
                  (unsigned)(NPROP - 1 - i);
    }
    __syncthreads();

    // ---- 4) Bitonic sort (descending) of 2048 64-bit keys in LDS ----
    for (int k = 2; k <= NPROP; k <<= 1) {
        for (int j = k >> 1; j > 0; j >>= 1) {
            for (int e = 0; e < 2; ++e) {
                const int i   = t + e * NTHREADS;
                const int ixj = i ^ j;
                if (ixj > i) {
                    const unsigned long long a  = sKey[i];
                    const unsigned long long bb = sKey[ixj];
                    const bool desc = (i & k) == 0;
                    if (desc ? (a < bb) : (a > bb)) { sKey[i] = bb; sKey[ixj] = a; }
                }
            }
            __syncthreads();
        }
    }

    // ---- 5) Greedy NMS: only kept candidates emit a suppression row ----
    for (;;) {
        if (t == 0) {
            int pick = -1;
            if (sCount < DETS) {
                int i = sCur;
                while (i < NPROP) {
                    const unsigned ou = (unsigned)(sKey[i] >> 32);
                    if (ou <= 0x80000000u) { i = NPROP; break; }   // score<=0 -> invalid tail
                    if (!((sSup[i >> 5] >> (i & 31)) & 1u)) { pick = i; break; }
                    ++i;
                }
                sCur = (pick >= 0) ? (pick + 1) : NPROP;
            }
            if (pick >= 0) {
                const unsigned long long kk = sKey[pick];
                const unsigned idx = (NPROP - 1) - (unsigned)(kk & 0xFFFFFFFFu);
                const int o = sCount;
                float* ob = outBoxes + ((size_t)b * DETS + o) * 4;
                ob[0] = sBox[idx * 4 + 0]; ob[1] = sBox[idx * 4 + 1];
                ob[2] = sBox[idx * 4 + 2]; ob[3] = sBox[idx * 4 + 3];
                outScores[b * DETS + o]  = ord2f((unsigned)(kk >> 32));
                outClasses[b * DETS + o] = 1;
                outValid[b * DETS + o]   = 1;
                sCount = o + 1;
            }
            sPick = pick;
        }
        __syncthreads();
        const int pick = sPick;
        if (pick < 0) break;

        const unsigned pidx = (NPROP - 1) - (unsigned)(sKey[pick] & 0xFFFFFFFFu);
        const float px1 = sBox[pidx * 4 + 0], py1 = sBox[pidx * 4 + 1];
        const float px2 = sBox[pidx * 4 + 2], py2 = sBox[pidx * 4 + 3];
        const float parea = (px2 - px1) * (py2 - py1);

        for (int e = 0; e < 2; ++e) {
            const int jj = t + e * NTHREADS;
            bool pred = false;
            if (jj > pick) {
                const unsigned jidx = (NPROP - 1) - (unsigned)(sKey[jj] & 0xFFFFFFFFu);
                const float jx1 = sBox[jidx * 4 + 0], jy1 = sBox[jidx * 4 + 1];
                const float jx2 = sBox[jidx * 4 + 2], jy2 = sBox[jidx * 4 + 3];
                const float ix1 = fmaxf(px1, jx1), iy1 = fmaxf(py1, jy1);
                const float ix2 = fminf(px2, jx2), iy2 = fminf(py2, jy2);
                const float iw = fmaxf(ix2 - ix1, 0.0f), ih = fmaxf(iy2 - iy1, 0.0f);
                const float inter = iw * ih;
                const float jarea = (jx2 - jx1) * (jy2 - jy1);
                pred = (inter / (parea + jarea - inter + 1e-9f)) > NMS_T;
            }
            // wave32: 32 lanes cover exactly one mask word of consecutive j's
            const unsigned m = ballot32(pred);
            if ((t & 31) == 0) sSup[jj >> 5] |= m;
        }
        __syncthreads();
    }

    // ---- 6) Zero the unfilled tail (harness poisons d_out) ----
    if (t == 0) {
        for (int o = sCount; o < DETS; ++o) {
            float* ob = outBoxes + ((size_t)b * DETS + o) * 4;
            ob[0] = ob[1] = ob[2] = ob[3] = 0.0f;
            outScores[b * DETS + o]  = 0.0f;
            outClasses[b * DETS + o] = 0;
            outValid[b * DETS + o]   = 0;
        }
    }
}

extern "C" void kernel_launch(void* const* d_in, const int* in_sizes, int n_in,
                              void* d_out, int out_size, void* d_ws, size_t ws_size,
                              hipStream_t stream) {
    (void)in_sizes; (void)n_in; (void)d_ws; (void)ws_size; (void)out_size;
    const float* cls   = (const float*)d_in[0];  // prop_clss [8,2048,2]
    const float* regs  = (const float*)d_in[1];  // prop_regs [8,2048,4]
    const float* qlt   = (const float*)d_in[2];  // prop_qlts [8,2048,1]
    const float* props = (const float*)d_in[3];  // props     [8,2048,4]

    // d_out: outputs concatenated flat in return order:
    //   boxes  f32  [8,100,4] -> 3200 floats
    //   scores f32  [8,100]   ->  800 floats
    //   classes i32 [8,100]   ->  800 int32 (bit-stored in the next 800 4-byte slots)
    //   valid  bool [8,100]   ->  800 bytes after that
    float* base      = (float*)d_out;
    float* outBoxes  = base;
    float* outScores = base + BIMG * DETS * 4;
    int*   outClasses = (int*)(base + BIMG * DETS * 4 + BIMG * DETS);
    unsigned char* outValid =
        (unsigned char*)(base + BIMG * DETS * 4 + 2 * BIMG * DETS);

    hipLaunchKernelGGL(SEAS_postprocess_kernel, dim3(BIMG), dim3(NTHREADS), 0, stream,
                       cls, regs, qlt, props, outBoxes, outScores, outClasses, outValid);
}